// SenderAgent_12575664243382
// MI455X (gfx1250) — compile-verified
//
#include <hip/hip_runtime.h>
#include <hip/hip_bf16.h>

#define N_NODES 100000
#define C 128

typedef __attribute__((ext_vector_type(2))) float v2f;
typedef __attribute__((ext_vector_type(8))) float v8f;

// ---------------- degree / normalization ----------------
__global__ void deg_init_kernel(unsigned int* __restrict__ deg, int n) {
    int i = blockIdx.x * blockDim.x + threadIdx.x;
    if (i < n) deg[i] = 1u;  // self-loop contributes 1 to every node's degree
}

__global__ void deg_count_kernel(const int* __restrict__ dst, unsigned int* __restrict__ deg, int e) {
    int i = blockIdx.x * blockDim.x + threadIdx.x;
    if (i < e) atomicAdd(&deg[dst[i]], 1u);   // native global_atomic_add_u32
}

__global__ void deg_rsqrt_kernel(float* __restrict__ dinv, const unsigned int* __restrict__ deg, int n) {
    int i = blockIdx.x * blockDim.x + threadIdx.x;
    if (i < n) dinv[i] = 1.0f / sqrtf((float)deg[i]);  // deg >= 1 always
}

// ---------------- GEMM: Y[M,128] = X[M,128] @ W[128,128] via V_WMMA_F32_16X16X4_F32 ----
// Block = 256 threads = 8 waves; wave w owns N-tile w (16 cols); block owns one 16-row M-tile.
// A 16x4 f32: lanes 0-15 hold row=lane, K={k,k+1}; lanes 16-31 hold row=lane-16, K={k+2,k+3}.
// B 4x16 f32: VGPR0 = K-row {k | k+2} striped over lane halves, VGPR1 = K-row {k+1 | k+3}.
// C/D 16x16 f32: VGPR i -> (M = i | i+8) for lane halves, N = lane&15.
__global__ void __launch_bounds__(256)
gemm_wmma_f32(const float* __restrict__ X, const float* __restrict__ W,
              float* __restrict__ Y) {
    const int lane = threadIdx.x & 31;
    const int wave = threadIdx.x >> 5;
    const int m0 = blockIdx.x << 4;
    const int n0 = wave << 4;
    const int row = lane & 15;
    const int kh  = (lane >> 4) << 1;  // 0 or 2

    v8f c = {};
    const float* xp = X + (size_t)(m0 + row) * C + kh;
    const float* wp = W + (size_t)kh * C + n0 + row;

#pragma unroll
    for (int k = 0; k < C; k += 4) {
        v2f a = *(const v2f*)(xp + k);           // X[m0+row][k+kh], [k+kh+1]
        v2f b;
        b.x = wp[(size_t)k * C];                 // W[k+kh  ][n0+row]
        b.y = wp[(size_t)(k + 1) * C];           // W[k+kh+1][n0+row]
        c = __builtin_amdgcn_wmma_f32_16x16x4_f32(
                false, a, false, b, (short)0, c, false, false);
    }

    const int rbase = m0 + ((lane >> 4) << 3);
    const int col = n0 + row;
#pragma unroll
    for (int i = 0; i < 8; ++i)
        Y[(size_t)(rbase + i) * C + col] = c[i];
}

// ---------------- self-loop init: acc[i,c] = xw[i,c] * dinv[i]^2 ----------------
__global__ void self_init_kernel(const float* __restrict__ xw, const float* __restrict__ dinv,
                                 float* __restrict__ acc, int total) {
    int idx = blockIdx.x * blockDim.x + threadIdx.x;
    if (idx < total) {
        float di = dinv[idx >> 7];
        acc[idx] = xw[idx] * di * di;
    }
}

// ---------------- edge scatter: wave-per-edge, lane handles 4 channels ----------------
__global__ void __launch_bounds__(256)
edge_scatter_kernel(const float* __restrict__ xw, float* __restrict__ acc,
                    const int* __restrict__ srcIdx, const int* __restrict__ dstIdx,
                    const float* __restrict__ dinv, int e) {
    int edge = blockIdx.x * 8 + (threadIdx.x >> 5);
    if (edge >= e) return;
    int lane = threadIdx.x & 31;
    int s = srcIdx[edge];
    int d = dstIdx[edge];
    float norm = dinv[s] * dinv[d];
    const float4 v = *(const float4*)(xw + (size_t)s * C + lane * 4);
    float* outp = acc + (size_t)d * C + lane * 4;
    // native no-return fp32 atomics (global_atomic_add_f32) on the hot path
    unsafeAtomicAdd(outp + 0, v.x * norm);
    unsafeAtomicAdd(outp + 1, v.y * norm);
    unsafeAtomicAdd(outp + 2, v.z * norm);
    unsafeAtomicAdd(outp + 3, v.w * norm);
}

// ---------------- bias + relu (in place) ----------------
__global__ void bias_relu_kernel(float* __restrict__ h, const float* __restrict__ bias, int total) {
    int idx = blockIdx.x * blockDim.x + threadIdx.x;
    if (idx < total) {
        float vv = h[idx] + bias[idx & (C - 1)];
        h[idx] = vv > 0.0f ? vv : 0.0f;
    }
}

// ---------------- mean pool ----------------
__global__ void zero_vec_kernel(float* __restrict__ g, int n) {
    int i = blockIdx.x * blockDim.x + threadIdx.x;
    if (i < n) g[i] = 0.0f;
}

__global__ void mean_pool_kernel(const float* __restrict__ h, float* __restrict__ g,
                                 int rows_per_block, float inv_n) {
    int cidx = threadIdx.x;                 // 128 threads = channels
    int r0 = blockIdx.x * rows_per_block;
    float s = 0.0f;
    for (int i = 0; i < rows_per_block; ++i)
        s += h[(size_t)(r0 + i) * C + cidx];
    unsafeAtomicAdd(&g[cidx], s * inv_n);
}

// ---------------- final FC: out[j] = sum_c (g[c]+b2[c]) * Wfc[c,j] + bfc[j] ----------------
__global__ void fc_kernel(const float* __restrict__ g, const float* __restrict__ b2,
                          const float* __restrict__ Wfc, const float* __restrict__ bfc,
                          float* __restrict__ out) {
    int j = threadIdx.x;  // 256 threads
    float s = bfc[j];
#pragma unroll 8
    for (int cc = 0; cc < C; ++cc)
        s += (g[cc] + b2[cc]) * Wfc[(size_t)cc * 256 + j];
    out[j] = s;
}

extern "C" void kernel_launch(void* const* d_in, const int* in_sizes, int n_in,
                              void* d_out, int out_size, void* d_ws, size_t ws_size,
                              hipStream_t stream) {
    const float* x   = (const float*)d_in[0];
    const int*   ei  = (const int*)d_in[1];
    const float* W1  = (const float*)d_in[2];
    const float* b1  = (const float*)d_in[3];
    const float* W2  = (const float*)d_in[4];
    const float* b2  = (const float*)d_in[5];
    const float* Wfc = (const float*)d_in[6];
    const float* bfc = (const float*)d_in[7];
    float* out = (float*)d_out;

    const int E = in_sizes[1] / 2;          // edge_index is [2, E]
    const int* src = ei;                    // row
    const int* dst = ei + E;                // col

    // workspace layout: bufA | bufB | dinv | degcnt | g   (~103 MB)
    float* bufA = (float*)d_ws;
    float* bufB = bufA + (size_t)N_NODES * C;
    float* dinv = bufB + (size_t)N_NODES * C;
    unsigned int* degcnt = (unsigned int*)(dinv + N_NODES);
    float* g    = (float*)(degcnt + N_NODES);

    const int total = N_NODES * C;
    const int tb = 256;
    const int gN   = (N_NODES + tb - 1) / tb;
    const int gE   = (E + tb - 1) / tb;
    const int gTot = (total + tb - 1) / tb;
    const int gEdgeWaves = (E + 7) / 8;     // wave-per-edge, 8 edges per block
    const int gGemm = N_NODES / 16;         // 6250 M-tiles (100000 % 16 == 0)

    // degrees -> dinv
    deg_init_kernel<<<gN, tb, 0, stream>>>(degcnt, N_NODES);
    deg_count_kernel<<<gE, tb, 0, stream>>>(dst, degcnt, E);
    deg_rsqrt_kernel<<<gN, tb, 0, stream>>>(dinv, degcnt, N_NODES);

    // ---- layer 1 ----
    gemm_wmma_f32<<<gGemm, 256, 0, stream>>>(x, W1, bufA);                 // xw1
    self_init_kernel<<<gTot, tb, 0, stream>>>(bufA, dinv, bufB, total);    // self-loops
    edge_scatter_kernel<<<gEdgeWaves, 256, 0, stream>>>(bufA, bufB, src, dst, dinv, E);
    bias_relu_kernel<<<gTot, tb, 0, stream>>>(bufB, b1, total);            // h1 in bufB

    // ---- layer 2 ----
    gemm_wmma_f32<<<gGemm, 256, 0, stream>>>(bufB, W2, bufA);              // xw2 (h1 dead after)
    self_init_kernel<<<gTot, tb, 0, stream>>>(bufA, dinv, bufB, total);
    edge_scatter_kernel<<<gEdgeWaves, 256, 0, stream>>>(bufA, bufB, src, dst, dinv, E);
    // no relu after layer 2; b2 folded into fc via mean of a constant

    // ---- mean pool + fc ----
    zero_vec_kernel<<<1, 128, 0, stream>>>(g, 128);
    mean_pool_kernel<<<250, 128, 0, stream>>>(bufB, g, N_NODES / 250, 1.0f / (float)N_NODES);
    fc_kernel<<<1, 256, 0, stream>>>(g, b2, Wfc, bfc, out);
}